// GeometricGNN_74423193305352
// MI455X (gfx1250) — compile-verified
//
#include <hip/hip_runtime.h>

#define DH     128
#define DIN    32
#define CONCAT 768
#define NLAYERS 5

typedef __attribute__((ext_vector_type(2))) float v2f;
typedef __attribute__((ext_vector_type(8))) float v8f;

__device__ __forceinline__ void atomic_add_f32(float* p, float v) {
    __hip_atomic_fetch_add(p, v, __ATOMIC_RELAXED, __HIP_MEMORY_SCOPE_AGENT);
}

// ---------------------------------------------------------------- utilities
__global__ void zero_f32(float* __restrict__ p, int n) {
    int i = blockIdx.x * blockDim.x + threadIdx.x;
    if (i < n) p[i] = 0.0f;
}

// agg = 2*h  (self-loop + (1+eps)*h, eps=0); n4 = element count / 4
__global__ void init_agg4(const float4* __restrict__ h, float4* __restrict__ agg, int n4) {
    int i = blockIdx.x * blockDim.x + threadIdx.x;
    if (i < n4) {
        float4 v = h[i];
        agg[i] = make_float4(2.0f * v.x, 2.0f * v.y, 2.0f * v.z, 2.0f * v.w);
    }
}

// ------------------------------------------------- edge scatter (mem-bound)
// one edge per wave32: lane l handles floats [4l,4l+4)
__global__ void scatter_edges(const float* __restrict__ h, const int* __restrict__ ei,
                              float* __restrict__ agg, int E) {
    int wave = threadIdx.x >> 5;
    int lane = threadIdx.x & 31;
    int e = blockIdx.x * 8 + wave;
    if (e >= E) return;
    int src = ei[e];
    int dst = ei[E + e];
    const float4 v = *(const float4*)(h + (size_t)src * DH + lane * 4);
    float* d = agg + (size_t)dst * DH + lane * 4;
    atomic_add_f32(d + 0, v.x);
    atomic_add_f32(d + 1, v.y);
    atomic_add_f32(d + 2, v.z);
    atomic_add_f32(d + 3, v.w);
}

// ------------------------------------------------- embed: h = x @ W_embed
// block: 16 rows, 8 waves; wave w computes the 16x16 tile at cols [16w,16w+16)
__global__ void embed_gemm(const float* __restrict__ x, const float* __restrict__ W,
                           float* __restrict__ h, int N) {
    __shared__ float As[16 * (DIN + 4)];  // stride 36 avoids bank conflicts
    const int tid  = threadIdx.x;
    const int row0 = blockIdx.x * 16;
    for (int i = tid; i < 16 * DIN; i += 256) {
        int r = i >> 5, c = i & (DIN - 1);
        int gr = row0 + r;
        As[r * (DIN + 4) + c] = (gr < N) ? x[(size_t)gr * DIN + c] : 0.0f;
    }
    __syncthreads();
    const int wave = tid >> 5, lane = tid & 31;
    const int n0 = wave * 16;
    const int m  = lane & 15, hi = lane >> 4;
    v8f acc = {};
#pragma unroll
    for (int k0 = 0; k0 < DIN; k0 += 4) {
        int ka = k0 + 2 * hi;
        v2f a, b;
        a.x = As[m * (DIN + 4) + ka];
        a.y = As[m * (DIN + 4) + ka + 1];
        b.x = W[(size_t)ka * DH + n0 + m];
        b.y = W[(size_t)(ka + 1) * DH + n0 + m];
        acc = __builtin_amdgcn_wmma_f32_16x16x4_f32(false, a, false, b,
                                                    (short)0, acc, false, false);
    }
    if (row0 + 16 <= N) {  // uniform fast path: straight stores, imm offsets
        float* dst = h + (size_t)(row0 + 8 * hi) * DH + n0 + m;
#pragma unroll
        for (int i = 0; i < 8; i++) dst[i * DH] = acc[i];
    } else {
#pragma unroll
        for (int i = 0; i < 8; i++) {
            int gr = row0 + i + 8 * hi;
            if (gr < N) h[(size_t)gr * DH + n0 + m] = acc[i];
        }
    }
}

// ------------------------------------- layer GEMM: h = agg @ W[l] + b[l]
__global__ void gin_gemm(const float* __restrict__ agg, const float* __restrict__ W,
                         const float* __restrict__ bias, float* __restrict__ h, int N) {
    __shared__ float As[16 * (DH + 4)];  // stride 132: lanes m=0..15 hit distinct banks
    const int tid  = threadIdx.x;
    const int row0 = blockIdx.x * 16;
    for (int i = tid; i < 16 * DH; i += 256) {
        int r = i >> 7, c = i & (DH - 1);
        int gr = row0 + r;
        As[r * (DH + 4) + c] = (gr < N) ? agg[(size_t)gr * DH + c] : 0.0f;
    }
    __syncthreads();
    const int wave = tid >> 5, lane = tid & 31;
    const int n0 = wave * 16;
    const int m  = lane & 15, hi = lane >> 4;
    v8f acc = {};
#pragma unroll 4
    for (int k0 = 0; k0 < DH; k0 += 4) {
        int ka = k0 + 2 * hi;
        v2f a, b;
        a.x = As[m * (DH + 4) + ka];
        a.y = As[m * (DH + 4) + ka + 1];
        b.x = W[(size_t)ka * DH + n0 + m];
        b.y = W[(size_t)(ka + 1) * DH + n0 + m];
        acc = __builtin_amdgcn_wmma_f32_16x16x4_f32(false, a, false, b,
                                                    (short)0, acc, false, false);
    }
    float bv = bias[n0 + m];
    if (row0 + 16 <= N) {  // uniform fast path
        float* dst = h + (size_t)(row0 + 8 * hi) * DH + n0 + m;
#pragma unroll
        for (int i = 0; i < 8; i++) dst[i * DH] = acc[i] + bv;
    } else {
#pragma unroll
        for (int i = 0; i < 8; i++) {
            int gr = row0 + i + 8 * hi;
            if (gr < N) h[(size_t)gr * DH + n0 + m] = acc[i] + bv;
        }
    }
}

// ----------------------------- graph pooling: out[batch[n], off+c] += h[n,c]
// one float4 per thread (32 float4 per node row)
__global__ void pool_scatter4(const float* __restrict__ h, const int* __restrict__ batch,
                              float* __restrict__ out, int off, int N) {
    int idx = blockIdx.x * blockDim.x + threadIdx.x;
    if (idx >= N * (DH / 4)) return;
    int n = idx >> 5;            // node
    int c = (idx & 31) * 4;      // column
    int g = batch[n];
    const float4 v = *(const float4*)(h + (size_t)n * DH + c);
    float* d = out + (size_t)g * CONCAT + off + c;
    atomic_add_f32(d + 0, v.x);
    atomic_add_f32(d + 1, v.y);
    atomic_add_f32(d + 2, v.z);
    atomic_add_f32(d + 3, v.w);
}

// ---------------------------------------------------------------- driver
extern "C" void kernel_launch(void* const* d_in, const int* in_sizes, int n_in,
                              void* d_out, int out_size, void* d_ws, size_t ws_size,
                              hipStream_t stream) {
    const float* x       = (const float*)d_in[0];
    const int*   edges   = (const int*)d_in[1];
    const int*   batch   = (const int*)d_in[2];
    const float* W_embed = (const float*)d_in[3];
    const float* Ws      = (const float*)d_in[4];
    const float* bs      = (const float*)d_in[5];
    float* out = (float*)d_out;

    const int N = in_sizes[0] / DIN;   // 50000
    const int E = in_sizes[1] / 2;     // 800000
    const size_t HN = (size_t)N * DH;  // 6.4M floats

    float* h   = (float*)d_ws;         // [N,128]  25.6 MB
    float* agg = h + HN;               // [N,128]  25.6 MB

    // output is atomically accumulated -> must be zeroed every launch
    zero_f32<<<(out_size + 255) / 256, 256, 0, stream>>>(out, out_size);

    const int gemm_blocks = (N + 15) / 16;
    const int hn4         = (int)(HN / 4);
    const int pool_blocks = (N * (DH / 4) + 255) / 256;

    embed_gemm<<<gemm_blocks, 256, 0, stream>>>(x, W_embed, h, N);
    pool_scatter4<<<pool_blocks, 256, 0, stream>>>(h, batch, out, 0, N);

    for (int l = 0; l < NLAYERS; l++) {
        init_agg4<<<(hn4 + 255) / 256, 256, 0, stream>>>(
            (const float4*)h, (float4*)agg, hn4);
        scatter_edges<<<(E + 7) / 8, 256, 0, stream>>>(h, edges, agg, E);
        gin_gemm<<<gemm_blocks, 256, 0, stream>>>(
            agg, Ws + (size_t)l * DH * DH, bs + (size_t)l * DH, h, N);
        pool_scatter4<<<pool_blocks, 256, 0, stream>>>(h, batch, out, (l + 1) * DH, N);
    }
}